// RNNTLoss__33973191311815
// MI455X (gfx1250) — compile-verified
//
#include <hip/hip_runtime.h>
#include <hip/hip_bf16.h>

// Problem constants (from reference): acts is (B, T, U+1, V)
#define BB 16
#define TT 200
#define UU 100
#define VV 256
#define U1 (UU + 1)
#define NROWS (BB * TT * U1)      // 323,200 rows of V=256
#define ROWSTRIDE 260             // LDS row stride in floats (bank-conflict-free WMMA A reads)

typedef __attribute__((ext_vector_type(2))) float v2f;
typedef __attribute__((ext_vector_type(8))) float v8f;

__device__ __forceinline__ float lae(float a, float b) {
    float m = fmaxf(a, b);
    if (!(m > -__builtin_inff())) return -__builtin_inff();
    return m + __logf(__expf(a - m) + __expf(b - m));
}

// ---------------------------------------------------------------------------
// Kernel 1: fused log_softmax + gather.
// Each wave (32 lanes) owns a tile of 16 consecutive rows (16x256 f32).
//  Pass A: coalesced b128 global loads -> LDS tile; per-row max via shfl.
//  Pass B: sum of exp(x - max) via V_WMMA_F32_16X16X4_F32 with B = ones:
//          D[m][n] = sum_k exp(A[m][k]) accumulated over K=256 in steps of 4.
//  Output: blank_lp[row] and emit_lp[b,t,u] (gathered at labels[b,u]).
// Block = 128 threads = 4 waves = 64 rows. Grid = 323200/64 = 5050 blocks.
// ---------------------------------------------------------------------------
__global__ __launch_bounds__(128) void rnnt_logsoftmax_wmma(
    const float* __restrict__ acts, const int* __restrict__ labels,
    float* __restrict__ blank_lp, float* __restrict__ emit_lp)
{
    __shared__ float lds[4][16 * ROWSTRIDE];   // 66,560 bytes

    const int lane = threadIdx.x & 31;
    const int wave = threadIdx.x >> 5;
    const int tile = blockIdx.x * 4 + wave;
    const int rowBase = tile * 16;
    float* L = lds[wave];

    float myMax = -__builtin_inff();   // ends holding max of row (lane & 15)

    // ---- Pass A: load 16 rows coalesced, stage to LDS, per-row max ----
    #pragma unroll 1
    for (int i = 0; i < 16; ++i) {
        const float* rowp = acts + (size_t)(rowBase + i) * VV + lane * 8;
        float4 v0 = *(const float4*)rowp;
        float4 v1 = *(const float4*)(rowp + 4);
        *(float4*)(L + i * ROWSTRIDE + lane * 8)     = v0;
        *(float4*)(L + i * ROWSTRIDE + lane * 8 + 4) = v1;
        float m = fmaxf(fmaxf(fmaxf(v0.x, v0.y), fmaxf(v0.z, v0.w)),
                        fmaxf(fmaxf(v1.x, v1.y), fmaxf(v1.z, v1.w)));
        #pragma unroll
        for (int off = 16; off >= 1; off >>= 1)
            m = fmaxf(m, __shfl_xor(m, off, 32));
        myMax = ((lane & 15) == i) ? m : myMax;
    }

    // ---- Pass B: WMMA row-sum of exp(x - max) ----
    // A fragment (16x4 f32): lane l -> row (l&15), K-pair at k0 + 2*(l>>4).
    const int r = lane & 15;
    const float* Lr = L + r * ROWSTRIDE + ((lane >> 4) << 1);

    v2f bones; bones[0] = 1.0f; bones[1] = 1.0f;   // B = ones (4x16)
    v8f c = {0.f, 0.f, 0.f, 0.f, 0.f, 0.f, 0.f, 0.f};

    #pragma unroll 4
    for (int k0 = 0; k0 < VV; k0 += 4) {
        v2f a;
        a[0] = __expf(Lr[k0]     - myMax);
        a[1] = __expf(Lr[k0 + 1] - myMax);
        c = __builtin_amdgcn_wmma_f32_16x16x4_f32(
                /*neg_a=*/false, a, /*neg_b=*/false, bones,
                /*c_mod=*/(short)0, c, /*reuse_a=*/false, /*reuse_b=*/false);
    }

    // Extract row sums: row m (m<8) -> c[m] on lanes 0..15; row m (>=8) -> c[m-8] on lanes 16..31.
    float mySum = 0.0f;
    #pragma unroll
    for (int j = 0; j < 8; ++j) {
        float s0 = __shfl(c[j], 0, 32);
        float s1 = __shfl(c[j], 16, 32);
        mySum = (r == j)     ? s0 : mySum;
        mySum = (r == j + 8) ? s1 : mySum;
    }

    // ---- Output: lanes 0..15 each finish one row ----
    if (lane < 16) {
        const int row  = rowBase + lane;
        const float logZ = myMax + __logf(mySum);
        const int u  = row % U1;
        const int bt = row / U1;
        const int t  = bt % TT;
        const int bb = bt / TT;
        blank_lp[row] = L[lane * ROWSTRIDE] - logZ;            // vocab index 0 = BLANK
        if (u < UU) {
            const int lab = labels[bb * UU + u];               // in [1, V)
            emit_lp[(bb * TT + t) * UU + u] = L[lane * ROWSTRIDE + lab] - logZ;
        }
    }
}

// ---------------------------------------------------------------------------
// Kernel 2: forward DP (alpha) per example, anti-diagonal wavefront.
// One workgroup per example; blank/emit tables for the example are staged
// into LDS (157 KB of the 320 KB per WGP), so each of the 299 serial
// diagonal steps touches only LDS.
//   alpha(t,u) = lae(alpha(t-1,u) + blank(t-1,u), alpha(t,u-1) + emit(t,u-1))
// ---------------------------------------------------------------------------
__global__ __launch_bounds__(128) void rnnt_alpha(
    const float* __restrict__ blank_lp, const float* __restrict__ emit_lp,
    const int* __restrict__ act_lens, const int* __restrict__ label_lens,
    float* __restrict__ nll_out)
{
    __shared__ float sb[TT * U1];      // 80,800 B
    __shared__ float se[TT * UU];      // 80,000 B
    __shared__ float Ad[2][U1];        // double-buffered diagonal
    __shared__ float saved;

    const int b   = blockIdx.x;
    const int tid = threadIdx.x;

    const float* gb = blank_lp + (size_t)b * TT * U1;
    const float* ge = emit_lp  + (size_t)b * TT * UU;
    for (int i = tid; i < TT * U1; i += 128) sb[i] = gb[i];
    for (int i = tid; i < TT * UU; i += 128) se[i] = ge[i];

    const int t_len = act_lens[b];
    const int u_len = label_lens[b];
    const int d_tgt = (t_len - 1) + u_len;

    if (tid <= UU) Ad[0][tid] = (tid == 0) ? 0.0f : -__builtin_inff();
    if (tid == 0 && d_tgt == 0) saved = 0.0f;
    __syncthreads();

    for (int d = 1; d <= TT - 1 + UU; ++d) {
        const float* Ap = Ad[(d - 1) & 1];
        float*       Ac = Ad[d & 1];
        if (tid <= UU) {
            const int u = tid;
            const int t = d - u;
            float v = -__builtin_inff();
            if (t >= 0 && t < TT) {
                float fb = -__builtin_inff(), fl = -__builtin_inff();
                if (t >= 1) fb = Ap[u]     + sb[(t - 1) * U1 + u];
                if (u >= 1) fl = Ap[u - 1] + se[t * UU + (u - 1)];
                v = lae(fb, fl);
            }
            Ac[u] = v;
            if (d == d_tgt && u == u_len) saved = v;
        }
        __syncthreads();
    }

    if (tid == 0) {
        const float ll = saved + sb[(t_len - 1) * U1 + u_len];
        nll_out[b] = -ll;
    }
}

// ---------------------------------------------------------------------------
// Kernel 3: loss = mean_b( nll[b] / label_len[b] )
// ---------------------------------------------------------------------------
__global__ void rnnt_reduce(const float* __restrict__ nll,
                            const int* __restrict__ label_lens,
                            float* __restrict__ out)
{
    const int lane = threadIdx.x;
    float v = 0.0f;
    if (lane < BB) v = nll[lane] / (float)label_lens[lane];
    #pragma unroll
    for (int off = 16; off >= 1; off >>= 1)
        v += __shfl_xor(v, off, 32);
    if (lane == 0) out[0] = v / (float)BB;
}

extern "C" void kernel_launch(void* const* d_in, const int* in_sizes, int n_in,
                              void* d_out, int out_size, void* d_ws, size_t ws_size,
                              hipStream_t stream) {
    const float* acts       = (const float*)d_in[0];   // (B, T, U+1, V) f32
    const int*   labels     = (const int*)d_in[1];     // (B, U) i32
    const int*   act_lens   = (const int*)d_in[2];     // (B,)  i32
    const int*   label_lens = (const int*)d_in[3];     // (B,)  i32
    float* out = (float*)d_out;

    float* blank_lp = (float*)d_ws;                    // B*T*(U+1) f32
    float* emit_lp  = blank_lp + (size_t)BB * TT * U1; // B*T*U f32
    float* nll      = emit_lp  + (size_t)BB * TT * UU; // B f32

    // 323,200 rows / 16 rows-per-wave / 4 waves-per-block = 5050 blocks
    rnnt_logsoftmax_wmma<<<NROWS / 64, 128, 0, stream>>>(acts, labels, blank_lp, emit_lp);
    rnnt_alpha<<<BB, 128, 0, stream>>>(blank_lp, emit_lp, act_lens, label_lens, nll);
    rnnt_reduce<<<1, 32, 0, stream>>>(nll, label_lens, out);
}